// SupervisedVICRegLoss_35003983463134
// MI455X (gfx1250) — compile-verified
//
#include <hip/hip_runtime.h>

typedef float v2f __attribute__((ext_vector_type(2)));
typedef float v8f __attribute__((ext_vector_type(8)));

#define N_ROWS 32768
#define DFEAT  256
#define NCLS   16
#define EPS    1e-4f
#define LAM    25.0f
#define MU     25.0f
#define NU     1.0f
#define ALPHA  50.0f

// workspace layout (float units)
#define WS_SUM_A   0
#define WS_SQ_A    4096
#define WS_SUM_B   8192
#define WS_SQ_B    12288
#define WS_MEAN_A  16384
#define WS_MEAN_B  20480
#define WS_COUNTS  24576   // int[16]
#define WS_CURSOR  24592   // int[16]
#define WS_OFFSETS 24608   // int[16]
#define WS_SCAL    24624   // float[16]: 0=inv,1=vl_a,2=vl_b,3=class,4=cov_a,5=cov_b
#define WS_IDX     24640   // int[N_ROWS]
#define ZERO_N     24640

__global__ __launch_bounds__(256) void zero_kernel(float* ws, int n) {
    int i = blockIdx.x * 256 + threadIdx.x;
    if (i < n) ws[i] = 0.0f;
}

// Per-class sum / sumsq for one view; optionally counts + invariance partial.
__global__ __launch_bounds__(256) void stats_kernel(
    const float* __restrict__ z, const float* __restrict__ zo,
    const int* __restrict__ labels,
    float* __restrict__ gsum, float* __restrict__ gsq,
    int* __restrict__ counts, float* __restrict__ inv_out, int do_extra)
{
    __shared__ float s_sum[NCLS * DFEAT];   // 16 KB
    __shared__ float s_sq[NCLS * DFEAT];    // 16 KB
    __shared__ int   s_cnt[NCLS];
    __shared__ float s_red[256];
    const int t = threadIdx.x;              // t == feature index d
    for (int i = t; i < NCLS * DFEAT; i += 256) { s_sum[i] = 0.0f; s_sq[i] = 0.0f; }
    if (t < NCLS) s_cnt[t] = 0;
    __syncthreads();

    const int rows_per_block = N_ROWS / gridDim.x;
    const int r0 = blockIdx.x * rows_per_block;
    float invp = 0.0f;
    for (int r = r0; r < r0 + rows_per_block; ++r) {
        float v = z[r * DFEAT + t];
        int   c = labels[r];
        s_sum[c * DFEAT + t] += v;          // thread t exclusively owns column t
        s_sq[c * DFEAT + t]  += v * v;
        if (do_extra) {
            float d = v - zo[r * DFEAT + t];
            invp += d * d;
            if (t == 0) s_cnt[c]++;
        }
    }
    __syncthreads();
    for (int i = t; i < NCLS * DFEAT; i += 256) {
        atomicAdd(&gsum[i], s_sum[i]);
        atomicAdd(&gsq[i],  s_sq[i]);
    }
    if (do_extra) {
        if (t < NCLS) atomicAdd(&counts[t], s_cnt[t]);
        s_red[t] = invp;
        __syncthreads();
        for (int s = 128; s > 0; s >>= 1) {
            if (t < s) s_red[t] += s_red[t + s];
            __syncthreads();
        }
        if (t == 0) atomicAdd(inv_out, s_red[0]);
    }
}

// Means, variance hinge, class-margin loss, prefix offsets. Single block of 256.
__global__ __launch_bounds__(256) void finalize_kernel(float* __restrict__ ws)
{
    __shared__ float s_m[NCLS * DFEAT];     // midpoint means, 16 KB
    __shared__ float s_red[256];
    const int t = threadIdx.x;
    float* sum_a = ws + WS_SUM_A;  float* sq_a = ws + WS_SQ_A;
    float* sum_b = ws + WS_SUM_B;  float* sq_b = ws + WS_SQ_B;
    float* mean_a = ws + WS_MEAN_A; float* mean_b = ws + WS_MEAN_B;
    int*   counts = (int*)(ws + WS_COUNTS);
    int*   cursor = (int*)(ws + WS_CURSOR);
    int*   offsets = (int*)(ws + WS_OFFSETS);
    float* scal = ws + WS_SCAL;

    float vla = 0.0f, vlb = 0.0f;
    for (int c = 0; c < NCLS; ++c) {
        float n  = (float)counts[c];
        float ma = sum_a[c * DFEAT + t] / n;
        float mb = sum_b[c * DFEAT + t] / n;
        mean_a[c * DFEAT + t] = ma;
        mean_b[c * DFEAT + t] = mb;
        float va = (sq_a[c * DFEAT + t] - n * ma * ma) / (n - 1.0f);
        float vb = (sq_b[c * DFEAT + t] - n * mb * mb) / (n - 1.0f);
        vla += fmaxf(0.0f, 1.0f - sqrtf(va + EPS));
        vlb += fmaxf(0.0f, 1.0f - sqrtf(vb + EPS));
        s_m[c * DFEAT + t] = 0.5f * (ma + mb);
    }
    s_red[t] = vla;
    __syncthreads();
    for (int s = 128; s > 0; s >>= 1) { if (t < s) s_red[t] += s_red[t + s]; __syncthreads(); }
    if (t == 0) scal[1] = s_red[0] / (float)(NCLS * DFEAT);
    __syncthreads();
    s_red[t] = vlb;
    __syncthreads();
    for (int s = 128; s > 0; s >>= 1) { if (t < s) s_red[t] += s_red[t + s]; __syncthreads(); }
    if (t == 0) scal[2] = s_red[0] / (float)(NCLS * DFEAT);

    if (t == 0) {
        int off = 0;
        for (int c = 0; c < NCLS; ++c) {
            offsets[c] = off;
            cursor[c]  = off;
            off += counts[c];
        }
    }
    __syncthreads();

    // class-discriminative margin loss over 120 pairs
    float pen = 0.0f;   // accumulated by thread 0
    for (int i = 0; i < NCLS; ++i) {
        for (int j = i + 1; j < NCLS; ++j) {
            float dd = s_m[i * DFEAT + t] - s_m[j * DFEAT + t];
            s_red[t] = dd * dd;
            __syncthreads();
            for (int s = 128; s > 0; s >>= 1) { if (t < s) s_red[t] += s_red[t + s]; __syncthreads(); }
            if (t == 0) {
                float dist = sqrtf(s_red[0]);
                float r = fmaxf(0.0f, ALPHA - dist);
                pen += r * r;
            }
            __syncthreads();
        }
    }
    if (t == 0) scal[3] = pen / (float)(NCLS * (NCLS - 1) / 2);
}

// Counting-sort row indices by class (order within a class is irrelevant).
__global__ __launch_bounds__(256) void scatter_kernel(
    const int* __restrict__ labels, int* __restrict__ cursor, int* __restrict__ idx)
{
    int i = blockIdx.x * 256 + threadIdx.x;
    if (i < N_ROWS) {
        int c = labels[i];
        int pos = atomicAdd(&cursor[c], 1);
        idx[pos] = i;
    }
}

// Per-class RAW Gram S_c via V_WMMA_F32_16X16X4_F32 with double-buffered
// GLOBAL_LOAD_ASYNC_TO_LDS_B128 staging; mean correction cov=(S-n*mu*mu^T)/(n-1)
// applied per-lane before the off-diagonal sum of squares.
// grid = 16 classes * 8 blocks; block = 8 waves; wave owns a 32x32 output region.
#define GCH        16                        // rows per chunk
#define BUF_FLOATS (GCH * DFEAT)             // 4096 floats = 16 KB
#define RED_BASE   (2 * BUF_FLOATS)          // reduction scratch after 2 buffers

__global__ __launch_bounds__(256) void gram_kernel(
    const float* __restrict__ z, const float* __restrict__ mean,
    const int* __restrict__ idx, const int* __restrict__ counts,
    const int* __restrict__ offsets, float* __restrict__ cov_out)
{
    // Single shared array => allocated at LDS offset 0; async asm uses raw offsets.
    __shared__ float s_all[RED_BASE + 256];  // 33 KB
    const int t    = threadIdx.x;
    const int cls  = blockIdx.x >> 3;
    const int blk  = blockIdx.x & 7;
    const int wave = t >> 5;
    const int lane = t & 31;
    const int reg  = blk * 8 + wave;         // 0..63 regions of 32x32 over 256x256
    const int rm   = (reg >> 3) * 32;
    const int rn   = (reg & 7) * 32;
    const int n_c  = counts[cls];
    const int off  = offsets[cls];

    v8f acc00 = {}, acc01 = {}, acc10 = {}, acc11 = {};
    const int mlow = lane & 15;
    const int kv   = (lane >> 4) * 2;        // lanes 0-15: K 0,1 ; lanes 16-31: K 2,3

    const int nfull = n_c >> 4;              // full 16-row chunks
    const int lrb   = t >> 6;                // this thread's base row-in-chunk
    const int c0    = (t & 63) * 4;          // this thread's starting column

    // Async-copy chunk `ch` of this class's rows into LDS buffer `buf`.
    auto issue = [&](int ch, int buf) {
        const int kbase = off + ch * GCH;
        const unsigned ldsbase = (unsigned)(buf * BUF_FLOATS * 4);
        #pragma unroll
        for (int i = 0; i < 4; ++i) {
            const int rlc = i * 4 + lrb;                 // row in chunk 0..15
            const int row = idx[kbase + rlc];
            const float* gp = z + (size_t)row * DFEAT + c0;
            const unsigned laddr = ldsbase + (unsigned)((rlc * DFEAT + c0) * 4);
            asm volatile("global_load_async_to_lds_b128 %0, %1, off"
                         :: "v"(laddr), "v"(gp) : "memory");
        }
    };
    auto compute = [&](int buf) {
        const float* sb = s_all + buf * BUF_FLOATS;
        #pragma unroll
        for (int kk = 0; kk < GCH; kk += 4) {
            const float* base0 = sb + (kk + kv) * DFEAT;
            const float* base1 = sb + (kk + kv + 1) * DFEAT;
            v2f a0, a1, b0, b1;
            a0.x = base0[rm + mlow];        a0.y = base1[rm + mlow];
            a1.x = base0[rm + 16 + mlow];   a1.y = base1[rm + 16 + mlow];
            b0.x = base0[rn + mlow];        b0.y = base1[rn + mlow];
            b1.x = base0[rn + 16 + mlow];   b1.y = base1[rn + 16 + mlow];
            acc00 = __builtin_amdgcn_wmma_f32_16x16x4_f32(false, a0, false, b0, (short)0, acc00, false, false);
            acc01 = __builtin_amdgcn_wmma_f32_16x16x4_f32(false, a0, false, b1, (short)0, acc01, false, false);
            acc10 = __builtin_amdgcn_wmma_f32_16x16x4_f32(false, a1, false, b0, (short)0, acc10, false, false);
            acc11 = __builtin_amdgcn_wmma_f32_16x16x4_f32(false, a1, false, b1, (short)0, acc11, false, false);
        }
    };

    if (nfull > 0) issue(0, 0);
    for (int c = 0; c < nfull; ++c) {
        const int cur = c & 1;
        if (c + 1 < nfull) {
            issue(c + 1, cur ^ 1);                       // prefetch next chunk
            asm volatile("s_wait_asynccnt 0x4" ::: "memory");  // oldest 4 (cur) done
        } else {
            asm volatile("s_wait_asynccnt 0x0" ::: "memory");
        }
        __syncthreads();
        compute(cur);
        __syncthreads();
    }

    const int rem = n_c & (GCH - 1);                     // tail rows via ds path
    if (rem) {
        for (int i = 0; i < GCH; ++i) {
            float v = 0.0f;
            if (i < rem) {
                int row = idx[off + nfull * GCH + i];
                v = z[(size_t)row * DFEAT + t];
            }
            s_all[i * DFEAT + t] = v;                    // zero rows contribute 0
        }
        __syncthreads();
        compute(0);
        __syncthreads();
    }

    // cov = (S - n*mu*mu^T)/(n-1); accumulate off-diagonal sum of squares
    const float nf = (float)n_c;
    const float* mu = mean + cls * DFEAT;
    const int mbase = (lane >> 4) * 8;                   // C/D: lanes16-31 hold M=8..15
    const int ncol  = lane & 15;
    const float mun0 = mu[rn + ncol];
    const float mun1 = mu[rn + 16 + ncol];
    float part = 0.0f;
    #pragma unroll
    for (int vg = 0; vg < 8; ++vg) {
        const int mg = rm + mbase + vg;
        const float mum0 = mu[mg];
        const float mum1 = mu[mg + 16];
        float x;
        x = acc00[vg] - nf * mum0 * mun0; if (mg      != rn + ncol     ) part += x * x;
        x = acc01[vg] - nf * mum0 * mun1; if (mg      != rn + 16 + ncol) part += x * x;
        x = acc10[vg] - nf * mum1 * mun0; if (mg + 16 != rn + ncol     ) part += x * x;
        x = acc11[vg] - nf * mum1 * mun1; if (mg + 16 != rn + 16 + ncol) part += x * x;
    }
    __syncthreads();
    float* s_red = s_all + RED_BASE;
    s_red[t] = part;
    __syncthreads();
    for (int s = 128; s > 0; s >>= 1) { if (t < s) s_red[t] += s_red[t + s]; __syncthreads(); }
    if (t == 0 && n_c >= 2) {
        float inv = 1.0f / (nf - 1.0f);
        atomicAdd(cov_out, s_red[0] * inv * inv);        // sum_{m!=n} cov^2 contribution
    }
}

__global__ void combine_kernel(const float* __restrict__ ws, float* __restrict__ out)
{
    const float* s = ws + WS_SCAL;
    float inv_loss = s[0] / (float)(N_ROWS * DFEAT);
    float var_loss = 0.5f * (s[1] + s[2]);
    float cov_loss = 0.5f * (s[4] + s[5]) / (float)(NCLS * DFEAT);
    float cls_loss = s[3];
    out[0] = LAM * inv_loss + MU * var_loss + NU * cov_loss + ALPHA * cls_loss;
}

extern "C" void kernel_launch(void* const* d_in, const int* in_sizes, int n_in,
                              void* d_out, int out_size, void* d_ws, size_t ws_size,
                              hipStream_t stream)
{
    const float* z_a    = (const float*)d_in[0];
    const float* z_b    = (const float*)d_in[1];
    const int*   labels = (const int*)d_in[2];
    float* ws  = (float*)d_ws;
    float* out = (float*)d_out;

    int*   counts  = (int*)(ws + WS_COUNTS);
    int*   cursor  = (int*)(ws + WS_CURSOR);
    int*   offsets = (int*)(ws + WS_OFFSETS);
    float* scal    = ws + WS_SCAL;
    int*   idx     = (int*)(ws + WS_IDX);

    zero_kernel<<<(ZERO_N + 255) / 256, 256, 0, stream>>>(ws, ZERO_N);
    stats_kernel<<<128, 256, 0, stream>>>(z_a, z_b, labels,
                                          ws + WS_SUM_A, ws + WS_SQ_A, counts, scal + 0, 1);
    stats_kernel<<<128, 256, 0, stream>>>(z_b, z_a, labels,
                                          ws + WS_SUM_B, ws + WS_SQ_B, counts, scal + 0, 0);
    finalize_kernel<<<1, 256, 0, stream>>>(ws);
    scatter_kernel<<<N_ROWS / 256, 256, 0, stream>>>(labels, cursor, idx);
    gram_kernel<<<NCLS * 8, 256, 0, stream>>>(z_a, ws + WS_MEAN_A, idx, counts, offsets, scal + 4);
    gram_kernel<<<NCLS * 8, 256, 0, stream>>>(z_b, ws + WS_MEAN_B, idx, counts, offsets, scal + 5);
    combine_kernel<<<1, 1, 0, stream>>>(ws, out);
}